// CrossAttentionMS_OneK_17145509446498
// MI455X (gfx1250) — compile-verified
//
#include <hip/hip_runtime.h>
#include <hip/hip_bf16.h>

typedef float v2f __attribute__((ext_vector_type(2)));
typedef float v8f __attribute__((ext_vector_type(8)));

constexpr int Bb = 8;
constexpr int Nn = 2048;
constexpr int Cc = 128;
constexpr int Hh = 4;
constexpr int Dd = 32;
constexpr int S1 = 3;
constexpr int Kk = 16;
constexpr int Kt = 48;   // K * S1

// ---------------------------------------------------------------------------
// Kernel 1: brute-force KNN top-48 (ascending distance, lower-index tiebreak)
// One 256-thread block per (b, n). Distance row lives in LDS; 48 sequential
// argmin passes give sorted neighbor indices (order matters for Wk slicing).
// ---------------------------------------------------------------------------
__global__ __launch_bounds__(256) void knn_kernel(const float* __restrict__ coord,
                                                  int* __restrict__ idxbuf) {
    __shared__ float sdist[Nn];
    __shared__ float wval[8];
    __shared__ int   widx[8];

    const int bn = blockIdx.x;
    const int b  = bn >> 11;          // N = 2048
    const int n  = bn & (Nn - 1);
    const int t  = threadIdx.x;

    const float* cb = coord + (size_t)b * Nn * 3;
    const float qx = cb[n * 3 + 0];
    const float qy = cb[n * 3 + 1];
    const float qz = cb[n * 3 + 2];

    for (int m = t; m < Nn; m += 256) {
        const float x = cb[m * 3 + 0];
        const float y = cb[m * 3 + 1];
        const float z = cb[m * 3 + 2];
        // |cm|^2 - 2*cq.cm : same ordering as full squared distance
        sdist[m] = (x * x + y * y + z * z) - 2.0f * (qx * x + qy * y + qz * z);
    }
    __syncthreads();

    const int lane = t & 31;
    const int wv   = t >> 5;

    for (int j = 0; j < Kt; ++j) {
        float best = 3.4e38f;
        int   bidx = 1 << 30;
        for (int m = t; m < Nn; m += 256) {
            const float s = sdist[m];
            if (s < best) { best = s; bidx = m; }   // strict < keeps lowest index
        }
        // wave32 reduction
        for (int off = 16; off > 0; off >>= 1) {
            const float ov = __shfl_down(best, (unsigned)off, 32);
            const int   oi = __shfl_down(bidx, (unsigned)off, 32);
            if (ov < best || (ov == best && oi < bidx)) { best = ov; bidx = oi; }
        }
        if (lane == 0) { wval[wv] = best; widx[wv] = bidx; }
        __syncthreads();
        if (t == 0) {
            float bv2 = wval[0];
            int   bi  = widx[0];
            for (int w2 = 1; w2 < 8; ++w2) {
                if (wval[w2] < bv2 || (wval[w2] == bv2 && widx[w2] < bi)) {
                    bv2 = wval[w2]; bi = widx[w2];
                }
            }
            sdist[bi] = 3.4e38f;                       // mask out for next pass
            idxbuf[(size_t)bn * Kt + j] = bi;
        }
        __syncthreads();
    }
}

// ---------------------------------------------------------------------------
// Kernel 2: q / v projections via V_WMMA_F32_16X16X4_F32.
// grid = (1024 row tiles, 2 {q,v}), block = 256 (8 waves); each wave owns one
// 16x16 output tile, K-loop of 32 fp32 WMMAs over C=128.
// ---------------------------------------------------------------------------
__global__ __launch_bounds__(256) void qv_kernel(const float* __restrict__ X,
                                                 const float* __restrict__ Wq,
                                                 const float* __restrict__ bq,
                                                 const float* __restrict__ Wv,
                                                 const float* __restrict__ bv,
                                                 float* __restrict__ qb,
                                                 float* __restrict__ vb) {
    const float* W    = blockIdx.y ? Wv : Wq;
    const float* bias = blockIdx.y ? bv : bq;
    float*       out  = blockIdx.y ? vb : qb;

    const int wave = threadIdx.x >> 5;
    const int lane = threadIdx.x & 31;
    const int row0 = blockIdx.x * 16;
    const int n0   = wave * 16;
    const int r    = lane & 15;
    const int kh   = lane >> 4;   // 0: K pair {0,1}, 1: K pair {2,3}

    v8f acc = {};
    for (int kk2 = 0; kk2 < 32; ++kk2) {
        const int k0 = kk2 * 4 + kh * 2;
        v2f a, bm;
        const float* xr = X + (size_t)(row0 + r) * Cc + k0;
        a.x  = xr[0];
        a.y  = xr[1];
        bm.x = W[(size_t)k0 * Cc + n0 + r];
        bm.y = W[(size_t)(k0 + 1) * Cc + n0 + r];
        acc = __builtin_amdgcn_wmma_f32_16x16x4_f32(false, a, false, bm,
                                                    (short)0, acc, false, false);
    }
#pragma unroll
    for (int j = 0; j < 8; ++j) {
        const int m   = j + kh * 8;
        const int col = n0 + r;
        out[(size_t)(row0 + m) * Cc + col] = acc[j] + bias[col];
    }
}

// ---------------------------------------------------------------------------
// Kernel 3: per-point gather + diff, K-projection (WMMA), logits, softmax,
// attn_map, output scaling. One 256-thread block per (b, n).
// ---------------------------------------------------------------------------
__global__ __launch_bounds__(256) void attn_kernel(const float* __restrict__ pcd,
                                                   const int* __restrict__ idxbuf,
                                                   const float* __restrict__ Wk,
                                                   const float* __restrict__ bk,
                                                   const float* __restrict__ qb,
                                                   const float* __restrict__ vb,
                                                   float* __restrict__ out) {
    __shared__ float diffS[Kt * Cc];      // 24 KB
    __shared__ float kS[Kt * Cc];         // 24 KB
    __shared__ int   idxS[Kt];
    __shared__ float centerS[Cc];
    __shared__ float qS[Cc];
    __shared__ float logitsS[S1 * Hh * Kk];
    __shared__ float attnSH[S1 * Hh];
    __shared__ float attnH[Hh];

    const int bn = blockIdx.x;
    const int b  = bn >> 11;
    const int t  = threadIdx.x;

    const float* pb   = pcd + (size_t)b * Nn * Cc;
    const float* prow = pcd + (size_t)bn * Cc;

    if (t < Kt)  idxS[t] = idxbuf[(size_t)bn * Kt + t];
    if (t < Cc) { centerS[t] = prow[t]; qS[t] = qb[(size_t)bn * Cc + t]; }
    __syncthreads();

    // Gather neighbors and form diff = nb - center  (48x128, uniform 24 iters)
    for (int e = t; e < Kt * Cc; e += 256) {
        const int j = e >> 7;
        const int c = e & 127;
        diffS[e] = pb[(size_t)idxS[j] * Cc + c] - centerS[c];
    }
    __syncthreads();

    // K projection: wave w computes cols [16w, 16w+16) for all 3 slices.
    const int wave = t >> 5;
    const int lane = t & 31;
    const int n0   = wave * 16;
    const int r    = lane & 15;
    const int kh   = lane >> 4;

    for (int s = 0; s < S1; ++s) {
        const float* Ws = Wk + (size_t)s * Cc * Cc;
        v8f acc = {};
        for (int kk2 = 0; kk2 < 32; ++kk2) {
            const int k0  = kk2 * 4 + kh * 2;
            const int row = s * 16 + r;
            v2f a, bm;
            a.x  = diffS[row * Cc + k0];
            a.y  = diffS[row * Cc + k0 + 1];
            bm.x = Ws[(size_t)k0 * Cc + n0 + r];
            bm.y = Ws[(size_t)(k0 + 1) * Cc + n0 + r];
            acc = __builtin_amdgcn_wmma_f32_16x16x4_f32(false, a, false, bm,
                                                        (short)0, acc, false, false);
        }
#pragma unroll
        for (int j = 0; j < 8; ++j) {
            const int m   = j + kh * 8;
            const int col = n0 + r;
            kS[(s * 16 + m) * Cc + col] = acc[j] + bk[s * Cc + col];
        }
    }
    __syncthreads();

    // logits[s][h][k] = (q[h,:] . k[s*16+k, h*32:h*32+32]) / sqrt(32)
    if (t < S1 * Hh * Kk) {
        const int s   = t >> 6;
        const int rem = t & 63;
        const int h   = rem >> 4;
        const int k   = rem & 15;
        const float* kr = &kS[(s * 16 + k) * Cc + h * Dd];
        const float* qr = &qS[h * Dd];
        float acc2 = 0.0f;
        for (int d = 0; d < Dd; ++d) acc2 += qr[d] * kr[d];
        logitsS[t] = acc2 * 0.17677669529663687f;   // 1/sqrt(32)
    }
    __syncthreads();

    // softmax over k, attn contribution = sum_k softmax_k * logit_k
    if (t < S1 * Hh) {
        const int s = t >> 2;
        const int h = t & 3;
        const float* lg = &logitsS[s * (Hh * Kk) + h * Kk];
        float mx = lg[0];
        for (int k = 1; k < Kk; ++k) mx = fmaxf(mx, lg[k]);
        float den = 0.0f, num = 0.0f;
        for (int k = 0; k < Kk; ++k) {
            const float e2 = expf(lg[k] - mx);
            den += e2;
            num += e2 * lg[k];
        }
        attnSH[t] = num / den;
    }
    __syncthreads();

    if (t < Hh) attnH[t] = attnSH[0 * Hh + t] + attnSH[1 * Hh + t] + attnSH[2 * Hh + t];
    __syncthreads();

    if (t < Cc) {
        const int h = t >> 5;
        out[(size_t)bn * Cc + t] = attnH[h] * vb[(size_t)bn * Cc + t];
    }
}

// ---------------------------------------------------------------------------
// Host launcher
// ---------------------------------------------------------------------------
extern "C" void kernel_launch(void* const* d_in, const int* in_sizes, int n_in,
                              void* d_out, int out_size, void* d_ws, size_t ws_size,
                              hipStream_t stream) {
    const float* pcd   = (const float*)d_in[0];
    const float* coord = (const float*)d_in[1];
    // d_in[2] is K (== 16), baked into Kt above
    const float* Wq = (const float*)d_in[3];
    const float* bq = (const float*)d_in[4];
    const float* Wk = (const float*)d_in[5];
    const float* bk = (const float*)d_in[6];
    const float* Wv = (const float*)d_in[7];
    const float* bv = (const float*)d_in[8];
    float* out = (float*)d_out;

    char* ws = (char*)d_ws;
    int*   idxbuf = (int*)ws;                                    // 16384*48 ints  (3 MB)
    float* qbuf   = (float*)(ws + (size_t)Bb * Nn * Kt * sizeof(int));
    float* vbuf   = qbuf + (size_t)Bb * Nn * Cc;                 // 8.4 MB each

    knn_kernel<<<Bb * Nn, 256, 0, stream>>>(coord, idxbuf);
    qv_kernel<<<dim3(Bb * Nn / 16, 2), 256, 0, stream>>>(pcd, Wq, bq, Wv, bv, qbuf, vbuf);
    attn_kernel<<<Bb * Nn, 256, 0, stream>>>(pcd, idxbuf, Wk, bk, qbuf, vbuf, out);
}